// MultiHeadAttnSerial_54975581389369
// MI455X (gfx1250) — compile-verified
//
#include <hip/hip_runtime.h>
#include <math.h>

// ---------------- problem constants ----------------
#define NHEADS   8
#define HEAD_DIM 64
#define HID      512          // NHEADS*HEAD_DIM
#define BATCH    32
#define MLEN     8192
#define NROWS    (BATCH * NHEADS)   // 256
#define SPLITS   8
#define CHUNK    (MLEN / SPLITS)    // 1024
#define TILE_ROWS 128               // key rows staged per async tile (32 KB)
#define NTILES   (CHUNK / TILE_ROWS)      // 8
#define RAMP_INV (1.0f / 32.0f)
#define SM_SCALE 0.125f             // 1/sqrt(64)

typedef __attribute__((ext_vector_type(2))) float v2f;
typedef __attribute__((ext_vector_type(8))) float v8f;

// ---------------- CDNA5 async-to-LDS helpers ----------------
typedef int v4i_vs __attribute__((vector_size(4 * sizeof(int))));
typedef __attribute__((address_space(1))) v4i_vs* as1_v4i;
typedef __attribute__((address_space(3))) v4i_vs* as3_v4i;

__device__ __forceinline__ void async_copy_b128(const float* __restrict__ g,
                                                float* l) {
#if __has_builtin(__builtin_amdgcn_global_load_async_to_lds_b128)
    __builtin_amdgcn_global_load_async_to_lds_b128(
        (as1_v4i)(void*)g, (as3_v4i)(void*)l,
        0 /*imm offset*/, 0 /*cpol*/);
#else
    // VDST = 32-bit LDS byte address (generic-pointer low bits per aperture
    // rule), VADDR = 64-bit global address.
    unsigned lds_off = (unsigned)(unsigned long long)(void*)l;
    asm volatile("global_load_async_to_lds_b128 %0, %1, off"
                 :: "v"(lds_off), "v"(g) : "memory");
#endif
}

#if __has_builtin(__builtin_amdgcn_s_wait_asynccnt)
#define WAIT_ASYNC(n) __builtin_amdgcn_s_wait_asynccnt(n)
#else
#define WAIT_ASYNC(n) asm volatile("s_wait_asynccnt %0" :: "i"(n) : "memory")
#endif

// =====================================================================
// Kernel 1/5: D[rows,cols] = A[rows,K] * W[cols,K]^T   (i.e. A @ W.T)
// One wave per 16x16 D tile; V_WMMA_F32_16X16X4_F32, K-loop step 4.
// =====================================================================
__global__ void wmma_gemm_nt(const float* __restrict__ A,
                             const float* __restrict__ W,
                             float* __restrict__ D,
                             int K, int ldA, int ldW, int ldD) {
    const int lane = threadIdx.x;
    const int half = lane >> 4;       // 0: lanes 0-15, 1: lanes 16-31
    const int lq   = lane & 15;
    const int cb   = blockIdx.x * 16; // column tile base
    const int rb   = blockIdx.y * 16; // row tile base

    const float* arow = A + (size_t)(rb + lq) * ldA;
    const float* wrow = W + (size_t)(cb + lq) * ldW;

    v8f c = {};
    for (int k = 0; k < K; k += 4) {
        const int ka = k + 2 * half;
        v2f a; a.x = arow[ka]; a.y = arow[ka + 1];
        v2f b; b.x = wrow[ka]; b.y = wrow[ka + 1];
        c = __builtin_amdgcn_wmma_f32_16x16x4_f32(
                false, a, false, b, (short)0, c, false, false);
    }
#pragma unroll
    for (int r = 0; r < 8; ++r) {
        const int mrow = r + 8 * half;
        D[(size_t)(rb + mrow) * ldD + cb + lq] = c[r];
    }
}

// =====================================================================
// Kernel 2: pe_bias[256,8192] = q[256,64] @ key_pe[64,8192]   (NN GEMM)
// =====================================================================
__global__ void pe_gemm(const float* __restrict__ q,
                        const float* __restrict__ key_pe,
                        float* __restrict__ peb) {
    const int lane = threadIdx.x;
    const int half = lane >> 4;
    const int lq   = lane & 15;
    const int cb   = blockIdx.x * 16;  // m tile
    const int rb   = blockIdx.y * 16;  // n tile

    const float* arow = q + (size_t)(rb + lq) * HEAD_DIM;

    v8f c = {};
    for (int k = 0; k < HEAD_DIM; k += 4) {
        const int ka = k + 2 * half;
        v2f a; a.x = arow[ka]; a.y = arow[ka + 1];
        v2f b;
        b.x = key_pe[(size_t)ka       * MLEN + cb + lq];
        b.y = key_pe[(size_t)(ka + 1) * MLEN + cb + lq];
        c = __builtin_amdgcn_wmma_f32_16x16x4_f32(
                false, a, false, b, (short)0, c, false, false);
    }
#pragma unroll
    for (int r = 0; r < 8; ++r) {
        const int mrow = r + 8 * half;
        peb[(size_t)(rb + mrow) * MLEN + cb + lq] = c[r];
    }
}

// =====================================================================
// Kernel 3: flash-decode partials. Grid (SPLITS, NROWS), 256 threads.
// Key tiles double-buffered in LDS via CDNA5 async-to-LDS copies.
// Tracks chunk max, Z = sum(exp), S = sum(exp*mask), acc = sum(exp*mask*v).
// =====================================================================
__global__ void attn_partial(const float* __restrict__ key,
                             const float* __restrict__ value,
                             const float* __restrict__ peb,
                             const float* __restrict__ qws,
                             const float* __restrict__ span,
                             float* __restrict__ pmax,
                             float* __restrict__ pz,
                             float* __restrict__ ps,
                             float* __restrict__ pacc) {
    __shared__ __align__(16) float kbuf[2][TILE_ROWS * HEAD_DIM]; // 2 x 32 KB
    __shared__ __align__(16) float q_s[HEAD_DIM];
    __shared__ float w_s[CHUNK];   // scores, then weights
    __shared__ float red[256];     // partial dots, then reductions

    const int split = blockIdx.x;
    const int n     = blockIdx.y;
    const int t     = threadIdx.x;
    const int m0    = split * CHUNK;

    const float* Kbase = key   + (size_t)n * MLEN * HEAD_DIM;
    const float* Vbase = value + (size_t)n * MLEN * HEAD_DIM;

    if (t < HEAD_DIM) q_s[t] = qws[n * HEAD_DIM + t];

    const float span_h = span[n & (NHEADS - 1)];

    // ---- prefetch tile 0 ----
    {
        const float* gsrc = Kbase + (size_t)m0 * HEAD_DIM;
        float*       ldst = &kbuf[0][0];
#pragma unroll
        for (int kq = 0; kq < 8; ++kq) {
            const int f4 = t + kq * 256;                 // float4 index in tile
            async_copy_b128(gsrc + f4 * 4, ldst + f4 * 4);
        }
    }

    // ---- pass 1: scores + local max, async double-buffered key tiles ----
    const int r  = t & (TILE_ROWS - 1);   // row within tile
    const int hh = t >> 7;                // 0/1: which half of d
    float lmax = -3.4e38f;

    for (int st = 0; st < NTILES; ++st) {
        if (st + 1 < NTILES) {
            const float* gsrc = Kbase + (size_t)(m0 + (st + 1) * TILE_ROWS) * HEAD_DIM;
            float*       ldst = &kbuf[(st + 1) & 1][0];
#pragma unroll
            for (int kq = 0; kq < 8; ++kq) {
                const int f4 = t + kq * 256;
                async_copy_b128(gsrc + f4 * 4, ldst + f4 * 4);
            }
            WAIT_ASYNC(8);    // tile st complete (in-order), st+1 in flight
        } else {
            WAIT_ASYNC(0);
        }
        __syncthreads();       // all waves' copies for tile st visible

        // two threads per row; swizzled float4 chunks to rotate LDS banks
        const float4* kb  = (const float4*)(&kbuf[st & 1][0]) + r * 16;
        const float4* qs4 = (const float4*)q_s;
        float part = 0.f;
#pragma unroll
        for (int j = 0; j < 8; ++j) {
            const int c = (r + 2 * j + hh) & 15;
            const float4 k4 = kb[c];
            const float4 q4 = qs4[c];
            part += k4.x * q4.x + k4.y * q4.y + k4.z * q4.z + k4.w * q4.w;
        }
        red[t] = part;
        __syncthreads();

        if (t < TILE_ROWS) {
            const int mm = st * TILE_ROWS + t;
            const int m  = m0 + mm;
            const float s = (red[t] + red[t + TILE_ROWS] +
                             peb[(size_t)n * MLEN + m]) * SM_SCALE;
            w_s[mm] = s;
            lmax = fmaxf(lmax, s);
        }
        // next iteration's top __syncthreads orders red[] reuse; the
        // __syncthreads after red[t]=part orders kbuf buffer reuse.
    }
    __syncthreads();

    red[t] = lmax;
    __syncthreads();
    for (int off = 128; off >= 1; off >>= 1) {
        if (t < off) red[t] = fmaxf(red[t], red[t + off]);
        __syncthreads();
    }
    const float gmax = red[0];
    __syncthreads();

    // ---- pass 2: exp + adaptive-span mask -> weights; Z and masked-S ----
    float z = 0.f, sm = 0.f;
#pragma unroll
    for (int j = 0; j < 4; ++j) {
        const int mm = j * 256 + t;
        const int m  = m0 + mm;
        const float e = __expf(w_s[mm] - gmax);
        float msk = ((float)(m - (MLEN - 1)) + span_h * (float)MLEN) * RAMP_INV + 1.0f;
        msk = fminf(fmaxf(msk, 0.f), 1.f);
        z += e;
        const float w = e * msk;
        sm += w;
        w_s[mm] = w;
    }
    red[t] = z;
    __syncthreads();
    for (int off = 128; off >= 1; off >>= 1) {
        if (t < off) red[t] += red[t + off];
        __syncthreads();
    }
    const float Z = red[0];
    __syncthreads();
    red[t] = sm;
    __syncthreads();
    for (int off = 128; off >= 1; off >>= 1) {
        if (t < off) red[t] += red[t + off];
        __syncthreads();
    }
    const float S = red[0];
    __syncthreads();

    // ---- pass 3: weighted value accumulation (coalesced 64-lane rows) ----
    const int d = t & 63;
    const int g = t >> 6;          // 4 m-groups of 256
    float acc = 0.f;
    const int mlo = g * 256, mhi = mlo + 256;
    for (int mm = mlo; mm < mhi; ++mm) {
        acc += w_s[mm] * Vbase[(size_t)(m0 + mm) * HEAD_DIM + d];
    }
    red[t] = acc;
    __syncthreads();

    const int idx = n * SPLITS + split;
    if (t < HEAD_DIM) {
        pacc[(size_t)idx * HEAD_DIM + t] =
            red[t] + red[t + 64] + red[t + 128] + red[t + 192];
    }
    if (t == 0) {
        pmax[idx] = gmax;
        pz[idx]   = Z;
        ps[idx]   = S;
    }
}

// =====================================================================
// Kernel 4: combine split partials, normalize: out = acc / (S + 1e-8*Z)
// =====================================================================
__global__ void attn_reduce(const float* __restrict__ pmax,
                            const float* __restrict__ pz,
                            const float* __restrict__ ps,
                            const float* __restrict__ pacc,
                            float* __restrict__ outh) {
    __shared__ float pm[SPLITS], zz[SPLITS], ss[SPLITS];
    const int n = blockIdx.x;
    const int d = threadIdx.x;
    if (d < SPLITS) {
        pm[d] = pmax[n * SPLITS + d];
        zz[d] = pz[n * SPLITS + d];
        ss[d] = ps[n * SPLITS + d];
    }
    __syncthreads();

    float gm = -3.4e38f;
#pragma unroll
    for (int s = 0; s < SPLITS; ++s) gm = fmaxf(gm, pm[s]);

    float Z = 0.f, S = 0.f, a = 0.f;
#pragma unroll
    for (int s = 0; s < SPLITS; ++s) {
        const float f = __expf(pm[s] - gm);
        Z += zz[s] * f;
        S += ss[s] * f;
        a += pacc[(size_t)(n * SPLITS + s) * HEAD_DIM + d] * f;
    }
    outh[n * HEAD_DIM + d] = a / (S + 1e-8f * Z);
}

// =====================================================================
// host launcher
// =====================================================================
extern "C" void kernel_launch(void* const* d_in, const int* in_sizes, int n_in,
                              void* d_out, int out_size, void* d_ws, size_t ws_size,
                              hipStream_t stream) {
    (void)in_sizes; (void)n_in; (void)out_size; (void)ws_size;
    const float* query  = (const float*)d_in[0];
    const float* key    = (const float*)d_in[1];
    const float* value  = (const float*)d_in[2];
    const float* Wq     = (const float*)d_in[3];
    const float* Wo     = (const float*)d_in[4];
    const float* key_pe = (const float*)d_in[5];
    const float* span   = (const float*)d_in[6];
    float* out = (float*)d_out;

    // workspace layout (floats); total ~2.27M floats (~9.1 MB)
    float* ws   = (float*)d_ws;
    float* qws  = ws;                           // 256*64      = 16384
    float* peb  = qws  + NROWS * HEAD_DIM;      // 256*8192    = 2097152
    float* pmax = peb  + (size_t)NROWS * MLEN;  // 256*8      = 2048
    float* pz   = pmax + NROWS * SPLITS;
    float* ps   = pz   + NROWS * SPLITS;
    float* pacc = ps   + NROWS * SPLITS;        // 256*8*64    = 131072
    float* outh = pacc + (size_t)NROWS * SPLITS * HEAD_DIM;   // 16384

    // 1) q = query @ Wq^T  -> qws viewed as [256,64]
    wmma_gemm_nt<<<dim3(HID / 16, BATCH / 16), 32, 0, stream>>>(
        query, Wq, qws, HID, HID, HID, HID);

    // 2) pe_bias = q @ key_pe  -> [256, 8192]
    pe_gemm<<<dim3(MLEN / 16, NROWS / 16), 32, 0, stream>>>(qws, key_pe, peb);

    // 3) flash-decode partials over 8 splits of M (async-staged key tiles)
    attn_partial<<<dim3(SPLITS, NROWS), 256, 0, stream>>>(
        key, value, peb, qws, span, pmax, pz, ps, pacc);

    // 4) combine partials -> head outputs [256,64] (== [32,512])
    attn_reduce<<<NROWS, HEAD_DIM, 0, stream>>>(pmax, pz, ps, pacc, outh);

    // 5) out = outh @ Wo^T -> [32,512]
    wmma_gemm_nt<<<dim3(HID / 16, BATCH / 16), 32, 0, stream>>>(
        outh, Wo, out, HID, HID, HID, HID);
}